// StdAttn_5188320493900
// MI455X (gfx1250) — compile-verified
//
#include <hip/hip_runtime.h>

// Problem constants (from reference): BSZ=32, NQ=2048, EMB=1024, ATT=512.
// softmax over a size-1 axis == 1.0, so out[b,e] = sum_nq v[b,nq,e].
// Only input v (d_in[2]) matters. 256 MB streamed once -> HBM-bound (~11us @ 23.3 TB/s).

#define BSZ 32
#define NQ  2048
#define EMB 1024
#define KSEG 4

typedef __attribute__((ext_vector_type(2))) float v2f;
typedef __attribute__((ext_vector_type(8))) float v8f;

// Column-sum of v[b, k0..k0+krows, colTile] via V_WMMA_F32_16X16X4_F32 with an
// all-ones A matrix: D[m,n] = sum_k B[k,n] + C[m,n]. Result is K-permutation
// invariant, so each lane-half just needs to cover 2 of the 4 rows per step.
// grid.x = b*8 + colgroup (128 cols each), grid.y = K segment.
// dst is indexed [blockIdx.y][b][col].
__global__ __launch_bounds__(256)
void colsum_wmma(const float* __restrict__ v, float* __restrict__ dst, int krows) {
    const int b    = blockIdx.x >> 3;        // 0..31
    const int cg   = blockIdx.x & 7;         // 0..7  (128-column groups)
    const int lane = threadIdx.x & 31;
    const int wave = threadIdx.x >> 5;       // 0..7  (16-column tile per wave)
    const int half = lane >> 4;              // 0 or 1
    const int col  = (cg << 7) + (wave << 4) + (lane & 15);
    const int k0   = blockIdx.y * krows;

    // lane-half 0 supplies rows k+0,k+1 ; lane-half 1 supplies rows k+2,k+3
    const float* p = v + (size_t)b * ((size_t)NQ * EMB)
                       + (size_t)(k0 + 2 * half) * EMB
                       + col;

    v2f a; a.x = 1.0f; a.y = 1.0f;           // all-ones A matrix (16x4)
    v8f acc[8];
#pragma unroll
    for (int u = 0; u < 8; ++u) acc[u] = (v8f){};

    // 32 rows per trip: 16 independent NT loads + 8 independent WMMAs
    for (int it = 0; it < krows; it += 32) {
        const float* pk = p + (size_t)it * EMB;
#pragma unroll
        for (int u = 0; u < 8; ++u) {
            v2f bb;
            bb.x = __builtin_nontemporal_load(pk + (size_t)(4 * u)     * EMB);
            bb.y = __builtin_nontemporal_load(pk + (size_t)(4 * u + 1) * EMB);
            acc[u] = __builtin_amdgcn_wmma_f32_16x16x4_f32(
                /*neg_a=*/false, a, /*neg_b=*/false, bb,
                /*c_mod=*/(short)0, acc[u], /*reuse_a=*/false, /*reuse_b=*/false);
        }
    }

    // Every (VGPR, lane) entry of each accumulator holds the column sum for
    // column N = lane%16 (all M rows identical since A is all ones).
    v8f s = acc[0] + acc[1] + acc[2] + acc[3] + acc[4] + acc[5] + acc[6] + acc[7];
    float r = s[0];

    if (lane < 16) {
        dst[(size_t)blockIdx.y * (BSZ * EMB) + (size_t)b * EMB + col] = r;
    }
}

// Sum the KSEG partial slabs into the final output (512 KB read, 128 KB write).
__global__ __launch_bounds__(256)
void finalize_sum(const float* __restrict__ ws, float* __restrict__ out) {
    int i = blockIdx.x * blockDim.x + threadIdx.x;
    if (i < BSZ * EMB) {
        float s = 0.0f;
#pragma unroll
        for (int t = 0; t < KSEG; ++t) s += ws[(size_t)t * (BSZ * EMB) + i];
        out[i] = s;
    }
}

extern "C" void kernel_launch(void* const* d_in, const int* in_sizes, int n_in,
                              void* d_out, int out_size, void* d_ws, size_t ws_size,
                              hipStream_t stream) {
    // inputs: 0=q 1=k 2=v 3=Wq 4=bq 5=Wk 6=bk 7=Ws 8=bs ; only v is live.
    const float* v = (const float*)d_in[2];
    float* out = (float*)d_out;

    const size_t need = (size_t)KSEG * BSZ * EMB * sizeof(float);
    if (d_ws != nullptr && ws_size >= need) {
        // 4-way K split: 1024 blocks (8192 waves) for latency hiding.
        dim3 grid(BSZ * 8, KSEG);
        colsum_wmma<<<grid, 256, 0, stream>>>(v, (float*)d_ws, NQ / KSEG);
        finalize_sum<<<(BSZ * EMB + 255) / 256, 256, 0, stream>>>((const float*)d_ws, out);
    } else {
        // Fallback: single pass straight into d_out.
        dim3 grid(BSZ * 8, 1);
        colsum_wmma<<<grid, 256, 0, stream>>>(v, out, NQ);
    }
}